// EPN_layer_48730698940570
// MI455X (gfx1250) — compile-verified
//
#include <hip/hip_runtime.h>

// EPN electron-passing layer for MI455X (gfx1250, wave32, WMMA).
// B=32, N=96, H=64, E=16, IN=146. All math f32 via V_WMMA_F32_16X16X4_F32.

#define BB 32
#define NN 96
#define HD 64

typedef __attribute__((ext_vector_type(2))) float v2f;
typedef __attribute__((ext_vector_type(8))) float v8f;

// ---------------------------------------------------------------------------
// Kernel 1: per-atom precompute
//   A[atom][c]  = x(atom) . W1[0:65, c]  + b1[c]     (x = [h(64), q])
//   Bm[atom][c] = x(atom) . W1[65:130, c]
// ---------------------------------------------------------------------------
__global__ __launch_bounds__(64) void epn_pre(const float* __restrict__ h,
                                              const float* __restrict__ q,
                                              const float* __restrict__ W1,
                                              const float* __restrict__ b1,
                                              float* __restrict__ Aws,
                                              float* __restrict__ Bws) {
  __shared__ float x[HD + 1];
  const int atom = blockIdx.x;
  const int t = threadIdx.x;
  if (t < HD) x[t] = h[(size_t)atom * HD + t];
  if (t == 0) x[HD] = q[atom];
  __syncthreads();
  const int col = t & 31;
  const int isB = t >> 5;
  const float* w = W1 + (isB ? 65 * 32 : 0) + col;
  float acc = isB ? 0.f : b1[col];
#pragma unroll
  for (int k = 0; k < HD + 1; ++k) acc += x[k] * w[k * 32];
  if (isB) Bws[(size_t)atom * 32 + col] = acc;
  else     Aws[(size_t)atom * 32 + col] = acc;
}

// ---------------------------------------------------------------------------
// Kernel 2: per-(b,i) row, one wave. For each 16-wide j tile:
//   Ec = e @ W1c               (WMMA f32 16x16x4, K=16 in 4 chunks, 2 n-tiles)
//   for s in {ij, ji}:
//     z1 = ReLU(Ec + A/Bm combo) -> LDS
//     z2 = ReLU(z1 @ W2 + b2)    (WMMA, K=32 in 8 chunks, 2 n-tiles) -> LDS
//     elec_s = z2 . W3 + b3      (lanes 0..15, one edge each)
//   acc += (elec_ij - elec_ji) * mask
// out[b,i] = q[b,i] + wave_reduce(acc)
// ---------------------------------------------------------------------------
__global__ __launch_bounds__(32) void epn_main(
    const float* __restrict__ e, const float* __restrict__ q,
    const float* __restrict__ mask,
    const float* __restrict__ W1, const float* __restrict__ W2,
    const float* __restrict__ b2, const float* __restrict__ W3,
    const float* __restrict__ b3,
    const float* __restrict__ Aws, const float* __restrict__ Bws,
    float* __restrict__ out) {
  __shared__ __align__(16) float zt[16][34];  // 16x32 tile, stride 34 (8B-aligned rows)
  __shared__ float w3s[32];

  const int row  = blockIdx.x;       // b*N + i
  const int b    = row / NN;
  const int lane = threadIdx.x;
  const int nloc = lane & 15;        // N (or M for A-operand) position
  const int hl   = lane >> 4;        // lane half -> K/M sub-block

  w3s[lane] = W3[lane];
  const float b3v = b3[0];

  // Weight operands in B-matrix layout (built once, reused for 6 j-tiles).
  // B 4x16 f32 layout: lane (n=nloc), VGPR0 = row K=2*hl, VGPR1 = row K=2*hl+1.
  v2f bW1c[4][2];   // W1 rows 130..145 : [16 x 32]
  v2f bW2c[8][2];   // W2 : [32 x 32]
  float ai[2], bmi[2], b2v[2];
#pragma unroll
  for (int t = 0; t < 2; ++t) {
    const int col = nloc + 16 * t;
#pragma unroll
    for (int c = 0; c < 4; ++c) {
      const int k = 4 * c + 2 * hl;
      bW1c[c][t].x = W1[(130 + k) * 32 + col];
      bW1c[c][t].y = W1[(131 + k) * 32 + col];
    }
#pragma unroll
    for (int c = 0; c < 8; ++c) {
      const int k = 4 * c + 2 * hl;
      bW2c[c][t].x = W2[k * 32 + col];
      bW2c[c][t].y = W2[(k + 1) * 32 + col];
    }
    ai[t]  = Aws[(size_t)row * 32 + col];   // A_i (includes b1)
    bmi[t] = Bws[(size_t)row * 32 + col];   // Bm_i
    b2v[t] = b2[col];
  }
  __syncthreads();

  float acc = 0.f;

  for (int jt = 0; jt < 6; ++jt) {
    const int j0 = jt * 16;

    // ---- shared edge term Ec = e @ W1c  (A 16x4: lane m=nloc, K=2hl,2hl+1)
    const float* erow = e + (((size_t)row) * NN + (j0 + nloc)) * 16;
    v8f ec0 = {};
    v8f ec1 = {};
#pragma unroll
    for (int c = 0; c < 4; ++c) {
      v2f a = *(const v2f*)(erow + 4 * c + 2 * hl);
      ec0 = __builtin_amdgcn_wmma_f32_16x16x4_f32(false, a, false, bW1c[c][0],
                                                  (short)0, ec0, false, false);
      ec1 = __builtin_amdgcn_wmma_f32_16x16x4_f32(false, a, false, bW1c[c][1],
                                                  (short)0, ec1, false, false);
    }

    // ---- per-j atom vectors (C/D layout: VGPR r holds row m = r + 8*hl)
    float aj[8][2], bmj[8][2];
#pragma unroll
    for (int r = 0; r < 8; ++r) {
      const int jatom = b * NN + j0 + r + 8 * hl;
#pragma unroll
      for (int t = 0; t < 2; ++t) {
        aj[r][t]  = Aws[(size_t)jatom * 32 + nloc + 16 * t];
        bmj[r][t] = Bws[(size_t)jatom * 32 + nloc + 16 * t];
      }
    }
    const float mval = mask[(size_t)row * NN + j0 + nloc];

    float elec[2];
#pragma unroll
    for (int s = 0; s < 2; ++s) {
      // z1 = ReLU(Ec + (s==0 ? A_i + Bm_j : A_j + Bm_i)) -> LDS
#pragma unroll
      for (int r = 0; r < 8; ++r) {
        const int m = r + 8 * hl;
        const float z0 = ec0[r] + (s == 0 ? ai[0] + bmj[r][0] : aj[r][0] + bmi[0]);
        const float z1 = ec1[r] + (s == 0 ? ai[1] + bmj[r][1] : aj[r][1] + bmi[1]);
        zt[m][nloc]      = fmaxf(z0, 0.f);
        zt[m][nloc + 16] = fmaxf(z1, 0.f);
      }
      __syncthreads();

      // layer 2: z2 = z1 @ W2  (A 16x4 from LDS, K=32 in 8 chunks)
      v8f d0 = {};
      v8f d1 = {};
#pragma unroll
      for (int c = 0; c < 8; ++c) {
        v2f a2 = *(const v2f*)&zt[nloc][4 * c + 2 * hl];
        d0 = __builtin_amdgcn_wmma_f32_16x16x4_f32(false, a2, false, bW2c[c][0],
                                                   (short)0, d0, false, false);
        d1 = __builtin_amdgcn_wmma_f32_16x16x4_f32(false, a2, false, bW2c[c][1],
                                                   (short)0, d1, false, false);
      }
      __syncthreads();  // all reads of zt done before overwrite

#pragma unroll
      for (int r = 0; r < 8; ++r) {
        const int m = r + 8 * hl;
        zt[m][nloc]      = fmaxf(d0[r] + b2v[0], 0.f);
        zt[m][nloc + 16] = fmaxf(d1[r] + b2v[1], 0.f);
      }
      __syncthreads();

      // layer 3: scalar dot per edge (lanes 0..15, edge m = lane)
      float el = b3v;
      if (lane < 16) {
#pragma unroll
        for (int n = 0; n < 32; ++n) el += zt[lane][n] * w3s[n];
      }
      elec[s] = el;
      __syncthreads();  // zt reused next s / next tile
    }

    if (lane < 16) acc += (elec[0] - elec[1]) * mval;
  }

  // reduce lanes 0..15 (lanes 16..31 hold 0)
#pragma unroll
  for (int off = 8; off > 0; off >>= 1) acc += __shfl_xor(acc, off, 32);
  if (lane == 0) out[row] = q[row] + acc;
}

// ---------------------------------------------------------------------------
extern "C" void kernel_launch(void* const* d_in, const int* in_sizes, int n_in,
                              void* d_out, int out_size, void* d_ws, size_t ws_size,
                              hipStream_t stream) {
  const float* h    = (const float*)d_in[0];
  const float* e    = (const float*)d_in[1];
  const float* q    = (const float*)d_in[2];
  const float* mask = (const float*)d_in[3];
  const float* W1   = (const float*)d_in[4];
  const float* b1   = (const float*)d_in[5];
  const float* W2   = (const float*)d_in[6];
  const float* b2   = (const float*)d_in[7];
  const float* W3   = (const float*)d_in[8];
  const float* b3   = (const float*)d_in[9];
  float* out_p = (float*)d_out;

  float* Aws = (float*)d_ws;                         // [B*N, 32]
  float* Bws = Aws + (size_t)BB * NN * 32;           // [B*N, 32]

  epn_pre<<<BB * NN, 64, 0, stream>>>(h, q, W1, b1, Aws, Bws);
  epn_main<<<BB * NN, 32, 0, stream>>>(e, q, mask, W1, W2, b2, W3, b3,
                                       Aws, Bws, out_p);
}